// EquivariantThreeHopGINE_61529701482729
// MI455X (gfx1250) — compile-verified
//
#include <hip/hip_runtime.h>
#include <math.h>

typedef float v2f __attribute__((ext_vector_type(2)));
typedef float v8f __attribute__((ext_vector_type(8)));

// ---------------------------------------------------------------------------
// Input index map: setup_inputs() dict insertion order, recursively flattened.
// ---------------------------------------------------------------------------
enum {
  IN_ATOM = 0,
  IN_ELEMENT, IN_DEGREE, IN_RING, IN_CHARGE, IN_AROMATIC, IN_HYBRID,
  IN_HYDROGEN, IN_FUNC_TABLES, IN_H_DON, IN_H_ACC, IN_RINGSIZE, IN_AROMA_NUM,
  IN_FUSED_ID, IN_FUNC_RED_W, IN_FUNC_RED_B, IN_BOND_ENV_W, IN_BOND_ENV_B,
  IN_LIN0_W, IN_LIN0_B, IN_LN_IN_G, IN_LN_IN_B, IN_BOND_EMB,
  IN_G1_EDGE_W, IN_G1_EDGE_B, IN_G1_W1, IN_G1_B1, IN_G1_W2, IN_G1_B2,
  IN_G2_EDGE_W, IN_G2_EDGE_B, IN_G2_W1, IN_G2_B1, IN_G2_W2, IN_G2_B2,
  IN_G3_EDGE_W, IN_G3_EDGE_B, IN_G3_W1, IN_G3_B1, IN_G3_W2, IN_G3_B2,
  IN_LN1_G, IN_LN1_B, IN_LN2_G, IN_LN2_B, IN_LN3_G, IN_LN3_B,
  IN_RES1, IN_RES2, IN_RES3,
  IN_MIX_W1, IN_MIX_B1, IN_MIX_W2, IN_MIX_B2, IN_OUT_W, IN_OUT_B,
  IN_PRE_G, IN_PRE_B,
  IN_SRC, IN_DST, IN_BOND_TYPE
};

#define N_NODES 100000
#define N_EDGES 400000

__device__ __forceinline__ int clampi(int v, int lo, int hi) {
  return v < lo ? lo : (v > hi ? hi : v);
}

// ---------------------------------------------------------------------------
// Precompute per-layer edge-type tables: ee[l][bt][j] = bond_emb'[bt] @ edge_w_l + edge_b_l
// (bond_emb row 0 zeroed per reference). 3*5*128 = 1920 outputs.
// ---------------------------------------------------------------------------
__global__ void ee_table_kernel(const float* __restrict__ bond_emb,
                                const float* __restrict__ ew0, const float* __restrict__ eb0,
                                const float* __restrict__ ew1, const float* __restrict__ eb1,
                                const float* __restrict__ ew2, const float* __restrict__ eb2,
                                float* __restrict__ ee) {
  int t = blockIdx.x * blockDim.x + threadIdx.x;
  if (t >= 3 * 5 * 128) return;
  int l = t / 640, rem = t % 640, b = rem / 128, j = rem % 128;
  const float* ew = (l == 0) ? ew0 : (l == 1) ? ew1 : ew2;
  const float* eb = (l == 0) ? eb0 : (l == 1) ? eb1 : eb2;
  float acc = eb[j];
  if (b != 0) {
    for (int k = 0; k < 32; ++k) acc += bond_emb[b * 32 + k] * ew[k * 128 + j];
  }
  ee[t] = acc;
}

// ---------------------------------------------------------------------------
// atom_embed: 78 raw cols -> 64 features per node (gathers + two tiny matvecs)
// ---------------------------------------------------------------------------
__global__ void atom_embed_kernel(const float* __restrict__ a,
    const float* __restrict__ emb_el, const float* __restrict__ deg,
    const float* __restrict__ ring,   const float* __restrict__ chg,
    const float* __restrict__ arom,   const float* __restrict__ hyb,
    const float* __restrict__ hyd,    const float* __restrict__ ftab,
    const float* __restrict__ hdon,   const float* __restrict__ hacc,
    const float* __restrict__ rsz,    const float* __restrict__ anum,
    const float* __restrict__ fid,    const float* __restrict__ frw,
    const float* __restrict__ frb,    const float* __restrict__ bew,
    const float* __restrict__ beb,
    float* __restrict__ feat, int n) {
  int i = blockIdx.x * blockDim.x + threadIdx.x;
  if (i >= n) return;
  const float* ar = a + (size_t)i * 78;
  int ai[30];
  #pragma unroll
  for (int j = 0; j < 30; ++j) ai[j] = (int)ar[j];  // trunc toward zero == astype(int32)

  const int ELUT[17] = {7,7,7,7,7,0,1,2,3,7,7,7,7,7,4,5,6};
  const int RLUT[9]  = {0,6,6,1,2,3,4,5,6};

  float f[64];
  int z = ai[0];
  int idx0 = (z < 0 || z > 16) ? 7 : ELUT[z];
  #pragma unroll
  for (int j = 0; j < 4; ++j) f[0  + j] = emb_el[idx0 * 4 + j];
  int i1 = clampi(ai[1], 0, 6);
  #pragma unroll
  for (int j = 0; j < 4; ++j) f[4  + j] = deg[i1 * 4 + j];
  int i2 = clampi(ai[5], 0, 1);
  #pragma unroll
  for (int j = 0; j < 4; ++j) f[8  + j] = ring[i2 * 4 + j];
  int i3 = clampi(ai[2], 0, 7);
  #pragma unroll
  for (int j = 0; j < 4; ++j) f[12 + j] = chg[i3 * 4 + j];
  int i4 = clampi(ai[4], 0, 1);
  #pragma unroll
  for (int j = 0; j < 4; ++j) f[16 + j] = arom[i4 * 4 + j];
  int i5 = clampi(ai[3], 0, 5);
  #pragma unroll
  for (int j = 0; j < 4; ++j) f[20 + j] = hyb[i5 * 4 + j];
  int i6 = clampi(ai[6], 0, 4);
  #pragma unroll
  for (int j = 0; j < 4; ++j) f[24 + j] = hyd[i6 * 4 + j];

  // flags (18 x 2 lookup -> 36) @ func_reduce_w (36,4) + b
  float flags[36];
  #pragma unroll
  for (int t = 0; t < 18; ++t) {
    int fx = clampi(ai[7 + t], 0, 1);
    flags[2 * t]     = ftab[t * 4 + fx * 2 + 0];
    flags[2 * t + 1] = ftab[t * 4 + fx * 2 + 1];
  }
  #pragma unroll
  for (int j = 0; j < 4; ++j) {
    float acc = frb[j];
    for (int k = 0; k < 36; ++k) acc += flags[k] * frw[k * 4 + j];
    f[28 + j] = acc;
  }
  int i25 = clampi(ai[25], 0, 1);
  f[32] = hdon[i25 * 2]; f[33] = hdon[i25 * 2 + 1];
  int i26 = clampi(ai[26], 0, 1);
  f[34] = hacc[i26 * 2]; f[35] = hacc[i26 * 2 + 1];
  int r27 = ai[27];
  int m27 = (r27 > 8) ? 6 : RLUT[clampi(r27, 0, 8)];
  #pragma unroll
  for (int j = 0; j < 4; ++j) f[36 + j] = rsz[m27 * 4 + j];
  int i28 = clampi(ai[28], 0, 4);
  #pragma unroll
  for (int j = 0; j < 4; ++j) f[40 + j] = anum[i28 * 4 + j];
  int i29 = clampi(ai[29], 0, 7);
  #pragma unroll
  for (int j = 0; j < 4; ++j) f[44 + j] = fid[i29 * 4 + j];

  // bond_env (48) @ bond_env_w (48,16) + b
  #pragma unroll
  for (int j = 0; j < 16; ++j) {
    float acc = beb[j];
    for (int k = 0; k < 48; ++k) acc += ar[30 + k] * bew[k * 16 + j];
    f[48 + j] = acc;
  }
  float* out = feat + (size_t)i * 64;
  #pragma unroll
  for (int j = 0; j < 64; ++j) out[j] = f[j];
}

// ---------------------------------------------------------------------------
// Row copy (initializes the segment-sum accumulator with h, so z = h + agg for free)
// ---------------------------------------------------------------------------
__global__ void copy_rows_kernel(const float* __restrict__ h, int ldh,
                                 float* __restrict__ z, int ldz, int n_rows) {
  int t = blockIdx.x * blockDim.x + threadIdx.x;
  int row = t >> 5;              // 32 threads x float4 = 128 cols
  if (row >= n_rows) return;
  int c = (t & 31) * 4;
  *(float4*)(z + (size_t)row * ldz + c) = *(const float4*)(h + (size_t)row * ldh + c);
}

// ---------------------------------------------------------------------------
// Edge messages: m = relu(h[src] + ee[bt]); atomic scatter-add into z[dst].
// One wave32 per edge (float4 per lane): gathers are 512B fully-coalesced rows.
// ---------------------------------------------------------------------------
__global__ void edge_scatter_kernel(const float* __restrict__ h, int ldh,
                                    const float* __restrict__ ee,     // [5][128]
                                    const int* __restrict__ src, const int* __restrict__ dst,
                                    const int* __restrict__ bt,
                                    float* __restrict__ z, int ldz, int n_edges) {
  int t = blockIdx.x * blockDim.x + threadIdx.x;
  int e = t >> 5;
  if (e >= n_edges) return;
  int c = (t & 31) * 4;
  int s = src[e], d = dst[e], b = bt[e];
  const float4 hv = *(const float4*)(h + (size_t)s * ldh + c);
  const float4 ev = *(const float4*)(ee + b * 128 + c);
  float m0 = fmaxf(hv.x + ev.x, 0.f);
  float m1 = fmaxf(hv.y + ev.y, 0.f);
  float m2 = fmaxf(hv.z + ev.z, 0.f);
  float m3 = fmaxf(hv.w + ev.w, 0.f);
  float* zp = z + (size_t)d * ldz + c;
  atomicAdd(zp + 0, m0);
  atomicAdd(zp + 1, m1);
  atomicAdd(zp + 2, m2);
  atomicAdd(zp + 3, m3);
}

// ---------------------------------------------------------------------------
// fp32 WMMA GEMM: C[16 x 128-tile] = act(A[16,K] @ W[K,M] + bias), optional fused
// LayerNorm (*g + b) and residual (+ res*resid) epilogue (LN requires M-tile==M==128).
// Block = 128 threads = 4 wave32s; each wave owns 2 16x16 accumulators (32 cols).
// Uses V_WMMA_F32_16X16X4_F32 (exact fp32 math, fp32 accumulate).
// ---------------------------------------------------------------------------
__global__ void __launch_bounds__(128)
wmma_gemm_f32(const float* __restrict__ A, int lda, int K,
              const float* __restrict__ W, int M,
              const float* __restrict__ bias,
              float* __restrict__ C, int ldc,
              int do_relu,
              const float* __restrict__ ln_g, const float* __restrict__ ln_b,
              const float* __restrict__ resid, int ldr,
              const float* __restrict__ res_scale,
              int n_rows) {
  __shared__ float As[16 * 514];   // K<=512, +2 pad to break LDS bank conflicts
  __shared__ float Cs[16 * 128];
  __shared__ float Ps[16 * 8], Ps2[16 * 8], Mu[16], Rsd[16];

  const int tid = threadIdx.x;
  const int lane = tid & 31;
  const int wid = tid >> 5;
  const int rowBase = blockIdx.x * 16;
  const int colBase = blockIdx.y * 128;
  const int lda_s = K + 2;

  // Stage 16 activation rows into LDS (zero-pad past n_rows)
  for (int r = 0; r < 16; ++r) {
    int grow = rowBase + r;
    for (int c = tid; c < K; c += 128)
      As[r * lda_s + c] = (grow < n_rows) ? A[(size_t)grow * lda + c] : 0.f;
  }
  __syncthreads();

  // WMMA fragment coordinates (16x16x4 f32 layout, wave32)
  const int m  = lane & 15;   // A row within tile / B col within 16-tile
  const int kp = lane >> 4;   // K parity half
  const int col0 = colBase + wid * 32 + m;

  v8f acc0 = {0.f, 0.f, 0.f, 0.f, 0.f, 0.f, 0.f, 0.f};
  v8f acc1 = {0.f, 0.f, 0.f, 0.f, 0.f, 0.f, 0.f, 0.f};

  for (int k0 = 0; k0 < K; k0 += 4) {
    int ka = k0 + 2 * kp;
    v2f a;
    a.x = As[m * lda_s + ka];
    a.y = As[m * lda_s + ka + 1];
    const float* w0 = W + (size_t)ka * M;
    v2f b0, b1;
    b0.x = w0[col0];           b0.y = w0[M + col0];
    b1.x = w0[col0 + 16];      b1.y = w0[M + col0 + 16];
    acc0 = __builtin_amdgcn_wmma_f32_16x16x4_f32(false, a, false, b0, (short)0, acc0, false, false);
    acc1 = __builtin_amdgcn_wmma_f32_16x16x4_f32(false, a, false, b1, (short)0, acc1, false, false);
  }

  float bb0 = bias ? bias[col0]      : 0.f;
  float bb1 = bias ? bias[col0 + 16] : 0.f;
  #pragma unroll
  for (int v = 0; v < 8; ++v) {
    acc0[v] += bb0;
    acc1[v] += bb1;
    if (do_relu) { acc0[v] = fmaxf(acc0[v], 0.f); acc1[v] = fmaxf(acc1[v], 0.f); }
  }

  if (ln_g) {
    // ---- fused LayerNorm (+ optional residual) over the 16x128 tile ----
    int cl = wid * 32 + m;  // colBase == 0 in LN mode
    #pragma unroll
    for (int v = 0; v < 8; ++v) {
      int r = v + 8 * kp;
      Cs[r * 128 + cl]      = acc0[v];
      Cs[r * 128 + cl + 16] = acc1[v];
    }
    __syncthreads();
    {
      int row = tid >> 3, seg = tid & 7;
      float s = 0.f, s2 = 0.f;
      #pragma unroll
      for (int c = 0; c < 16; ++c) {
        float x = Cs[row * 128 + seg * 16 + c];
        s += x; s2 += x * x;
      }
      Ps[row * 8 + seg] = s;
      Ps2[row * 8 + seg] = s2;
    }
    __syncthreads();
    if (tid < 16) {
      float s = 0.f, s2 = 0.f;
      #pragma unroll
      for (int k = 0; k < 8; ++k) { s += Ps[tid * 8 + k]; s2 += Ps2[tid * 8 + k]; }
      float mu = s * (1.f / 128.f);
      float var = s2 * (1.f / 128.f) - mu * mu;
      Mu[tid] = mu;
      Rsd[tid] = rsqrtf(var + 1e-5f);
    }
    __syncthreads();
    {
      int row = tid >> 3, seg = tid & 7;
      int grow = rowBase + row;
      if (grow < n_rows) {
        float mu = Mu[row], rs = Rsd[row];
        float rsc = res_scale ? res_scale[0] : 0.f;
        for (int c = seg * 16; c < seg * 16 + 16; ++c) {
          float y = (Cs[row * 128 + c] - mu) * rs * ln_g[c] + ln_b[c];
          if (resid) y += rsc * resid[(size_t)grow * ldr + c];
          C[(size_t)grow * ldc + c] = y;
        }
      }
    }
  } else {
    #pragma unroll
    for (int v = 0; v < 8; ++v) {
      int grow = rowBase + v + 8 * kp;
      if (grow < n_rows) {
        C[(size_t)grow * ldc + col0]      = acc0[v];
        C[(size_t)grow * ldc + col0 + 16] = acc1[v];
      }
    }
  }
}

// ---------------------------------------------------------------------------
// Host orchestration
// ---------------------------------------------------------------------------
extern "C" void kernel_launch(void* const* d_in, const int* in_sizes, int n_in,
                              void* d_out, int out_size, void* d_ws, size_t ws_size,
                              hipStream_t stream) {
  (void)in_sizes; (void)n_in; (void)out_size; (void)ws_size;
  const int N = N_NODES, E = N_EDGES;
  auto F = [&](int i) { return (const float*)d_in[i]; };
  const int* src = (const int*)d_in[IN_SRC];
  const int* dst = (const int*)d_in[IN_DST];
  const int* bti = (const int*)d_in[IN_BOND_TYPE];

  // workspace layout (fp32): ee(2048) | feat N*64 | H N*512 | Z N*128 | T1 N*128 | M1 N*256
  float* ws   = (float*)d_ws;
  float* ee   = ws;
  float* feat = ee + 2048;
  float* H    = feat + (size_t)N * 64;
  float* Z    = H + (size_t)N * 512;
  float* T1   = Z + (size_t)N * 128;
  float* M1   = T1 + (size_t)N * 128;
  float* out  = (float*)d_out;

  // 1) per-layer edge-type tables (5 distinct edge feature rows per layer)
  ee_table_kernel<<<(1920 + 127) / 128, 128, 0, stream>>>(
      F(IN_BOND_EMB),
      F(IN_G1_EDGE_W), F(IN_G1_EDGE_B),
      F(IN_G2_EDGE_W), F(IN_G2_EDGE_B),
      F(IN_G3_EDGE_W), F(IN_G3_EDGE_B), ee);

  // 2) atom embedding -> 64 features
  atom_embed_kernel<<<(N + 127) / 128, 128, 0, stream>>>(
      F(IN_ATOM),
      F(IN_ELEMENT), F(IN_DEGREE), F(IN_RING), F(IN_CHARGE), F(IN_AROMATIC),
      F(IN_HYBRID), F(IN_HYDROGEN), F(IN_FUNC_TABLES), F(IN_H_DON), F(IN_H_ACC),
      F(IN_RINGSIZE), F(IN_AROMA_NUM), F(IN_FUSED_ID), F(IN_FUNC_RED_W),
      F(IN_FUNC_RED_B), F(IN_BOND_ENV_W), F(IN_BOND_ENV_B), feat, N);

  dim3 g128((N + 15) / 16, 1);

  // 3) h0 = LN(feat @ lin0_w + b)  -> H[:, 0:128]  (row stride 512 => jk concat is free)
  wmma_gemm_f32<<<g128, 128, 0, stream>>>(
      feat, 64, 64, F(IN_LIN0_W), 128, F(IN_LIN0_B), H, 512,
      /*relu*/0, F(IN_LN_IN_G), F(IN_LN_IN_B), nullptr, 0, nullptr, N);

  // 4) three GINE layers
  const int LW1[3] = {IN_G1_W1, IN_G2_W1, IN_G3_W1};
  const int LB1[3] = {IN_G1_B1, IN_G2_B1, IN_G3_B1};
  const int LW2[3] = {IN_G1_W2, IN_G2_W2, IN_G3_W2};
  const int LB2[3] = {IN_G1_B2, IN_G2_B2, IN_G3_B2};
  const int LLG[3] = {IN_LN1_G, IN_LN2_G, IN_LN3_G};
  const int LLB[3] = {IN_LN1_B, IN_LN2_B, IN_LN3_B};
  const int LRS[3] = {IN_RES1, IN_RES2, IN_RES3};

  for (int l = 0; l < 3; ++l) {
    const float* hin = H + 128 * l;
    // z := h  (accumulator init), then z += relu(h[src] + ee[bt]) scattered to dst
    copy_rows_kernel<<<(N * 32 + 255) / 256, 256, 0, stream>>>(hin, 512, Z, 128, N);
    edge_scatter_kernel<<<(E * 32 + 255) / 256, 256, 0, stream>>>(
        hin, 512, ee + l * 640, src, dst, bti, Z, 128, E);
    // t1 = relu(z @ w1 + b1)
    wmma_gemm_f32<<<g128, 128, 0, stream>>>(
        Z, 128, 128, F(LW1[l]), 128, F(LB1[l]), T1, 128,
        /*relu*/1, nullptr, nullptr, nullptr, 0, nullptr, N);
    // h_{l+1} = LN(relu(t1 @ w2 + b2)) * g + b + res * h_l   -> H[:, 128*(l+1) : ...]
    wmma_gemm_f32<<<g128, 128, 0, stream>>>(
        T1, 128, 128, F(LW2[l]), 128, F(LB2[l]), H + 128 * (l + 1), 512,
        /*relu*/1, F(LLG[l]), F(LLB[l]), hin, 512, F(LRS[l]), N);
  }

  // 5) mix MLP on jk = H[:, 0:512]
  dim3 g256((N + 15) / 16, 2);
  wmma_gemm_f32<<<g256, 128, 0, stream>>>(
      H, 512, 512, F(IN_MIX_W1), 256, F(IN_MIX_B1), M1, 256,
      /*relu*/1, nullptr, nullptr, nullptr, 0, nullptr, N);
  wmma_gemm_f32<<<g128, 128, 0, stream>>>(
      M1, 256, 256, F(IN_MIX_W2), 128, F(IN_MIX_B2), Z, 128,
      /*relu*/1, nullptr, nullptr, nullptr, 0, nullptr, N);
  // 6) out = LN(Z @ out_w + out_b)
  wmma_gemm_f32<<<g128, 128, 0, stream>>>(
      Z, 128, 128, F(IN_OUT_W), 128, F(IN_OUT_B), out, 128,
      /*relu*/0, F(IN_PRE_G), F(IN_PRE_B), nullptr, 0, nullptr, N);
}